// Attention_58385785422060
// MI455X (gfx1250) — compile-verified
//
#include <hip/hip_runtime.h>
#include <math.h>

#define BB 32
#define SS 4096
#define HH 1024
#define NCHUNK 16
#define WPW 8                    // waves per workgroup
#define WPB (NCHUNK * WPW)       // 128 waves per batch
#define ROWS_PER_WAVE (SS / WPB) // 32

typedef __attribute__((ext_vector_type(2))) float v2f;
typedef __attribute__((ext_vector_type(8))) float v8f;

// ---------------------------------------------------------------------------
// Kernel 1: query = h_current @ W_a^T  via V_WMMA_F32_16X16X4_F32.
// One wave per 16-wide N tile; M=32 handled as two 16x16 accumulators.
// A (16x4 f32) lane layout: lane l -> row m=l&15, K-half (l>>4)*2, .x/.y = K,K+1
// B (4x16 f32) lane layout: lane l -> col n=l&15, K-half (l>>4)*2 (mirrors A)
// B[k][n] = W_a[n][k], so per-lane B loads are contiguous float2 of W_a row n.
// ---------------------------------------------------------------------------
__global__ __launch_bounds__(32)
void qgemm_wmma(const float* __restrict__ h, const float* __restrict__ W,
                float* __restrict__ q) {
    const int l     = threadIdx.x;
    const int n0    = blockIdx.x * 16;
    const int mrow  = l & 15;
    const int khalf = (l >> 4) * 2;
    const int ncol  = n0 + (l & 15);

    v8f c0 = {};
    v8f c1 = {};

    const float* arow0 = h + (size_t)mrow * HH + khalf;
    const float* arow1 = h + (size_t)(mrow + 16) * HH + khalf;
    const float* bcol  = W + (size_t)ncol * HH + khalf;

#pragma unroll 4
    for (int k0 = 0; k0 < HH; k0 += 4) {
        v2f a0 = *(const v2f*)(arow0 + k0);
        v2f a1 = *(const v2f*)(arow1 + k0);
        v2f bv = *(const v2f*)(bcol + k0);
        c0 = __builtin_amdgcn_wmma_f32_16x16x4_f32(false, a0, false, bv,
                                                   (short)0, c0, false, false);
        c1 = __builtin_amdgcn_wmma_f32_16x16x4_f32(false, a1, false, bv,
                                                   (short)0, c1, false, false);
    }

    // C/D layout: VGPR r: lanes0-15 -> M=r, lanes16-31 -> M=8+r
    const int mbase = (l >> 4) * 8;
#pragma unroll
    for (int r = 0; r < 8; ++r) {
        q[(size_t)(mbase + r) * HH + ncol]      = c0[r];
        q[(size_t)(16 + mbase + r) * HH + ncol] = c1[r];
    }
}

// ---------------------------------------------------------------------------
// Kernel 2: fused scores + online-softmax + context accumulation.
// Single read of all_hidden (512 MB): each row is used for its score AND its
// context contribution while resident in registers. Per-wave flash-style
// (m, l, c[1024]) state; deterministic ordered LDS merge per workgroup.
// Lane h-ownership: h = 4*lane + 128*j + comp, j=0..7 (matches float4 loads).
// ---------------------------------------------------------------------------
__global__ __launch_bounds__(256)
void attn_pass1(const float* __restrict__ allh, const float* __restrict__ query,
                float* __restrict__ scores, float* __restrict__ part_ml,
                float* __restrict__ part_c) {
    const int b     = blockIdx.x;
    const int chunk = blockIdx.y;
    const int lane  = threadIdx.x & 31;
    const int wave  = threadIdx.x >> 5;
    const int wid   = chunk * WPW + wave; // 0..127 within batch

    // Per-lane query slice (32 floats)
    const float* qb = query + (size_t)b * HH;
    float4 q[8];
#pragma unroll
    for (int j = 0; j < 8; ++j)
        q[j] = *(const float4*)(qb + lane * 4 + j * 128);

    float4 c[8];
#pragma unroll
    for (int j = 0; j < 8; ++j) c[j] = make_float4(0.f, 0.f, 0.f, 0.f);
    float m = -INFINITY, lsum = 0.f;

    const float* base = allh + (size_t)b * SS * HH;

    for (int t = 0; t < ROWS_PER_WAVE; ++t) {
        const int s = wid + t * WPB;
        const float* row = base + (size_t)s * HH;
        if (t + 1 < ROWS_PER_WAVE)
            __builtin_prefetch(base + (size_t)(s + WPB) * HH + lane * 16, 0, 0);

        float4 r[8];
        float part = 0.f;
#pragma unroll
        for (int j = 0; j < 8; ++j) {
            r[j] = *(const float4*)(row + lane * 4 + j * 128);
            part += r[j].x * q[j].x + r[j].y * q[j].y +
                    r[j].z * q[j].z + r[j].w * q[j].w;
        }
#pragma unroll
        for (int off = 16; off > 0; off >>= 1)
            part += __shfl_xor(part, off, 32);
        // 'part' is now the full dot product, uniform across the wave.
        if (lane == 0) scores[(size_t)b * SS + s] = part;

        float w;
        if (part > m) { // wave-uniform branch; rare after warm-up
            const float scale = __expf(m - part); // exp(-inf)=0 on first row
            lsum *= scale;
#pragma unroll
            for (int j = 0; j < 8; ++j) {
                c[j].x *= scale; c[j].y *= scale;
                c[j].z *= scale; c[j].w *= scale;
            }
            m = part;
            w = 1.f;
        } else {
            w = __expf(part - m);
        }
        lsum += w;
#pragma unroll
        for (int j = 0; j < 8; ++j) {
            c[j].x += w * r[j].x; c[j].y += w * r[j].y;
            c[j].z += w * r[j].z; c[j].w += w * r[j].w;
        }
    }

    // ---- deterministic workgroup merge of 8 wave partials ----
    __shared__ float lds_c[HH];
    __shared__ float lds_m[WPW];
    __shared__ float lds_l[WPW];
#pragma unroll
    for (int i = 0; i < 4; ++i) lds_c[threadIdx.x * 4 + i] = 0.f;
    if (lane == 0) { lds_m[wave] = m; lds_l[wave] = lsum; }
    __syncthreads();

    float Mwg = -INFINITY;
    for (int i = 0; i < WPW; ++i) Mwg = fmaxf(Mwg, lds_m[i]);
    float Lwg = 0.f;
    for (int i = 0; i < WPW; ++i) Lwg += lds_l[i] * __expf(lds_m[i] - Mwg);

    const float wscale = __expf(m - Mwg);
    for (int w8 = 0; w8 < WPW; ++w8) { // ordered, race-free, deterministic
        if (wave == w8) {
#pragma unroll
            for (int j = 0; j < 8; ++j) {
                const int hb = lane * 4 + j * 128;
                lds_c[hb]     += wscale * c[j].x;
                lds_c[hb + 1] += wscale * c[j].y;
                lds_c[hb + 2] += wscale * c[j].z;
                lds_c[hb + 3] += wscale * c[j].w;
            }
        }
        __syncthreads();
    }

    const int pidx = b * NCHUNK + chunk;
    if (threadIdx.x == 0) {
        part_ml[pidx * 2]     = Mwg;
        part_ml[pidx * 2 + 1] = Lwg;
    }
    float* pc = part_c + (size_t)pidx * HH;
#pragma unroll
    for (int i = 0; i < 4; ++i)
        pc[threadIdx.x * 4 + i] = lds_c[threadIdx.x * 4 + i];
}

// ---------------------------------------------------------------------------
// Kernel 3: combine 16 partials per batch -> context; normalize attn weights.
// Tiny: ~2.5 MB of reads total.
// ---------------------------------------------------------------------------
__global__ __launch_bounds__(256)
void attn_pass2(const float* __restrict__ scores, const float* __restrict__ part_ml,
                const float* __restrict__ part_c, float* __restrict__ ctx_out,
                float* __restrict__ attn_out) {
    const int b = blockIdx.x;

    float mi[NCHUNK], li[NCHUNK];
    float M = -INFINITY;
    for (int i = 0; i < NCHUNK; ++i) {
        mi[i] = part_ml[(b * NCHUNK + i) * 2];
        li[i] = part_ml[(b * NCHUNK + i) * 2 + 1];
        M = fmaxf(M, mi[i]);
    }
    float L = 0.f;
    for (int i = 0; i < NCHUNK; ++i) L += li[i] * __expf(mi[i] - M);
    const float invL = 1.f / L;

    // context: thread t owns h = 4t..4t+3
    const int hb = threadIdx.x * 4;
    float ax = 0.f, ay = 0.f, az = 0.f, aw = 0.f;
    for (int i = 0; i < NCHUNK; ++i) {
        const float sc = __expf(mi[i] - M);
        const float4 v = *(const float4*)(part_c + (size_t)(b * NCHUNK + i) * HH + hb);
        ax += sc * v.x; ay += sc * v.y; az += sc * v.z; aw += sc * v.w;
    }
    *(float4*)(ctx_out + (size_t)b * HH + hb) =
        make_float4(ax * invL, ay * invL, az * invL, aw * invL);

    // attention weights
    for (int s = threadIdx.x; s < SS; s += 256) {
        attn_out[(size_t)b * SS + s] =
            __expf(scores[(size_t)b * SS + s] - M) * invL;
    }
}

// ---------------------------------------------------------------------------
// Workspace layout (floats):
//   [0,       32768 )  query   [B,H]
//   [32768,  163840 )  scores  [B,S]
//   [163840, 164864 )  part_ml [B*NCHUNK*2]
//   [164864, 689152 )  part_c  [B*NCHUNK*H]     (total ~2.63 MB)
// Output: d_out = context [B,H] ++ attn_weights [B,S]
// ---------------------------------------------------------------------------
extern "C" void kernel_launch(void* const* d_in, const int* in_sizes, int n_in,
                              void* d_out, int out_size, void* d_ws, size_t ws_size,
                              hipStream_t stream) {
    const float* h_current  = (const float*)d_in[0];
    const float* all_hidden = (const float*)d_in[1];
    const float* W_a        = (const float*)d_in[2];
    float* out = (float*)d_out;
    float* ws  = (float*)d_ws;

    float* query   = ws;
    float* scores  = ws + 32768;
    float* part_ml = ws + 163840;
    float* part_c  = ws + 164864;

    qgemm_wmma<<<dim3(HH / 16), dim3(32), 0, stream>>>(h_current, W_a, query);
    attn_pass1<<<dim3(BB, NCHUNK), dim3(256), 0, stream>>>(
        all_hidden, query, scores, part_ml, part_c);
    attn_pass2<<<dim3(BB), dim3(256), 0, stream>>>(
        scores, part_ml, part_c, out, out + (size_t)BB * HH);
}